// SinkhornDistance_30983894073805
// MI455X (gfx1250) — compile-verified
//
#include <hip/hip_runtime.h>
#include <math.h>

// Sinkhorn distance on MI455X (gfx1250), wave32.
// Strategy:
//  - C = xn_i + yn_j - 2 x@y^T via V_WMMA_F32_16X16X4_F32 (fp32 matrix path),
//    K = exp(-lambda*C) written into the P-slot of d_out (regenerated every call).
//  - 100 gated Sinkhorn iterations, all kernels early-exit on device `done` flag.
//  - All reductions via fixed-order partial arrays (no float atomics -> bitwise
//    deterministic across graph replays).
//  - Final pass rewrites P in place and reduces cost into d_out[0].

#define NN 4096
#define MMC 4096
#define DD 64
#define LAMDA 0.01f
#define STOP_THR 1e-7f
#define MAX_ITER 100

typedef __attribute__((ext_vector_type(2))) float v2f;
typedef __attribute__((ext_vector_type(8))) float v8f;

// workspace layout (float offsets)
#define WS_A       0          // a        [4096]
#define WS_B       4096       // b        [4096]
#define WS_AN      8192       // a_new    [4096]
#define WS_BN      12288      // b_new    [4096]
#define WS_COLP    16384      // colpart  [32][4096]
#define WS_ROWP    147456     // rowpart  [4][4096]
#define WS_DISPP   163840     // disppart [2048]
#define WS_COSTP   165888     // costpart [2048]
#define WS_DONE    167936     // int done flag

__global__ void k_init(float* ws, float* out0) {
    int i = blockIdx.x * blockDim.x + threadIdx.x;
    if (i < NN) {
        ws[WS_A + i] = 1.0f / (float)NN;   // a0 = 1/n after normalization
        ws[WS_B + i] = 1.0f;               // b0 = 1
    }
    if (i == 0) {
        ((int*)(ws + WS_DONE))[0] = 0;
        out0[0] = 0.0f;
    }
}

// One wave -> one 16x16 tile of C/K. 16 chained V_WMMA_F32_16X16X4_F32 over D=64.
// Row norms of x/y are fused into the same float2 loads.
__global__ void __launch_bounds__(256)
k_cost_gemm(const float* __restrict__ x, const float* __restrict__ y,
            float* __restrict__ Cmat, float* __restrict__ Kmat) {
    const int lane = threadIdx.x & 31;
    const int wave = threadIdx.x >> 5;
    const int tN = blockIdx.x * 8 + wave;   // 0..255 (N tiles)
    const int tM = blockIdx.y;              // 0..255 (M tiles)
    const int r  = lane & 15;
    const int hi = lane >> 4;

    // A layout (16x4 f32): lanes 0-15 hold K=k0,k0+1; lanes 16-31 hold K=k0+2,k0+3.
    const float* xrow = x + (size_t)(tM * 16 + r) * DD + 2 * hi;
    const float* yrow = y + (size_t)(tN * 16 + r) * DD + 2 * hi;

    v8f acc = {};
    float xs = 0.f, ys = 0.f;   // fused row-norm partials (this lane's K slices)
#pragma unroll
    for (int k0 = 0; k0 < DD; k0 += 4) {
        v2f av = *(const v2f*)(xrow + k0);
        v2f bv = *(const v2f*)(yrow + k0);
        xs += av.x * av.x + av.y * av.y;
        ys += bv.x * bv.x + bv.y * bv.y;
        acc = __builtin_amdgcn_wmma_f32_16x16x4_f32(
            /*neg_a=*/false, av, /*neg_b=*/false, bv,
            /*c_mod=*/(short)0, acc, /*reuse_a=*/false, /*reuse_b=*/false);
    }
    // combine the two K-halves of each row's norm
    xs += __shfl_xor(xs, 16, 32);
    ys += __shfl_xor(ys, 16, 32);

    // C/D layout: VGPR v, lanes 0-15 -> row M0+v, lanes 16-31 -> row M0+8+v.
#pragma unroll
    for (int v = 0; v < 8; ++v) {
        int row = tM * 16 + v + 8 * hi;
        int col = tN * 16 + r;
        float xn = __shfl(xs, v + 8 * hi, 32);   // lane m holds xn of row M0+m
        float c  = xn + ys - 2.0f * acc[v];      // ys: own lane holds yn of its col
        size_t idx = (size_t)row * MMC + col;
        Cmat[idx] = c;
        Kmat[idx] = __expf(-LAMDA * c);
    }
}

// colpart[rc][j] = sum over 128-row chunk rc of K[i][j]*a[i]
__global__ void __launch_bounds__(256)
k_colsum(const float* __restrict__ Kmat, const float* ws, float* colpart,
         const int* done) {
    if (*done) return;
    int j  = blockIdx.x * 256 + threadIdx.x;
    int rc = blockIdx.y;                        // 0..31
    const float* a = ws + WS_A + rc * 128;
    const float* Kp = Kmat + (size_t)rc * 128 * MMC + j;
    float s = 0.f;
    for (int i = 0; i < 128; ++i)
        s += Kp[(size_t)i * MMC] * a[i];
    colpart[rc * MMC + j] = s;
}

// b_new[j] = nu[j] / sum_c colpart[c][j]  (fixed-order reduction)
__global__ void k_bnew(const float* __restrict__ nu, float* ws, const int* done) {
    if (*done) return;
    int j = blockIdx.x * blockDim.x + threadIdx.x;
    const float* cp = ws + WS_COLP;
    float s = 0.f;
#pragma unroll
    for (int c = 0; c < 32; ++c) s += cp[c * MMC + j];
    ws[WS_BN + j] = nu[j] / s;
}

// rowpart[cc][i] = sum over 1024-col chunk cc of K[i][j]*b_new[j]; one wave/row
__global__ void __launch_bounds__(256)
k_rowsum(const float* __restrict__ Kmat, const float* ws, float* rowpart,
         const int* done) {
    if (*done) return;
    int lane = threadIdx.x & 31;
    int wave = threadIdx.x >> 5;
    int row  = blockIdx.y * 8 + wave;
    int c0   = blockIdx.x * 1024;
    const float* bn = ws + WS_BN + c0;
    const float* Kp = Kmat + (size_t)row * MMC + c0;
    float s = 0.f;
#pragma unroll 8
    for (int c = lane; c < 1024; c += 32)
        s += Kp[c] * bn[c];
#pragma unroll
    for (int o = 16; o > 0; o >>= 1) s += __shfl_xor(s, o, 32);
    if (lane == 0) rowpart[blockIdx.x * NN + row] = s;
}

// a_new[i] = mu[i] / sum_c rowpart[c][i]
__global__ void k_anew(const float* __restrict__ mu, float* ws, const int* done) {
    if (*done) return;
    int i = blockIdx.x * blockDim.x + threadIdx.x;
    const float* rp = ws + WS_ROWP;
    float s = rp[i] + rp[NN + i] + rp[2 * NN + i] + rp[3 * NN + i];
    ws[WS_AN + i] = mu[i] / s;
}

// disppart[bid] = block partial of sum((a K b^T - a' K b'^T)^2)
__global__ void __launch_bounds__(256)
k_disp(const float* __restrict__ Kmat, const float* ws, float* disppart,
       const int* done, int first) {
    if (*done) return;
    const float* a  = ws + WS_A;
    const float* b  = ws + WS_B;
    const float* an = ws + WS_AN;
    const float* bn = ws + WS_BN;
    size_t base = (size_t)blockIdx.x * 8192;
    float s = 0.f;
    for (int t = threadIdx.x; t < 8192; t += 256) {
        size_t e = base + t;
        int i = (int)(e >> 12);
        int j = (int)(e & 4095);
        float k  = Kmat[e];
        float pn = an[i] * k * bn[j];
        float pp = first ? 0.f : a[i] * k * b[j];
        float d  = pp - pn;
        s += d * d;
    }
    __shared__ float red[256];
    red[threadIdx.x] = s;
    __syncthreads();
    for (int o = 128; o > 0; o >>= 1) {
        if (threadIdx.x < o) red[threadIdx.x] += red[threadIdx.x + o];
        __syncthreads();
    }
    if (threadIdx.x == 0) disppart[blockIdx.x] = red[0];
}

// gated by OLD done (matches reference: a_out uses done_{t-1})
__global__ void k_copy(float* ws, const int* done) {
    if (*done) return;
    int i = blockIdx.x * blockDim.x + threadIdx.x;
    ws[WS_A + i] = ws[WS_AN + i];
    ws[WS_B + i] = ws[WS_BN + i];
}

// reduce disppart -> disp, update done (single block, fixed order)
__global__ void k_done(const float* ws_disppart, int* done) {
    if (*done) return;
    __shared__ float red[256];
    float s = 0.f;
    for (int t = threadIdx.x; t < 2048; t += 256) s += ws_disppart[t];
    red[threadIdx.x] = s;
    __syncthreads();
    for (int o = 128; o > 0; o >>= 1) {
        if (threadIdx.x < o) red[threadIdx.x] += red[threadIdx.x + o];
        __syncthreads();
    }
    if (threadIdx.x == 0 && red[0] <= STOP_THR) *done = 1;
}

// P = a*K*b^T in place (K currently lives in the P slot); cost partials
__global__ void __launch_bounds__(256)
k_final(float* __restrict__ Pmat, const float* __restrict__ Cmat,
        const float* ws, float* costpart) {
    const float* a = ws + WS_A;
    const float* b = ws + WS_B;
    size_t base = (size_t)blockIdx.x * 8192;
    float s = 0.f;
    for (int t = threadIdx.x; t < 8192; t += 256) {
        size_t e = base + t;
        int i = (int)(e >> 12);
        int j = (int)(e & 4095);
        float p = a[i] * Pmat[e] * b[j];
        Pmat[e] = p;
        s += p * Cmat[e];
    }
    __shared__ float red[256];
    red[threadIdx.x] = s;
    __syncthreads();
    for (int o = 128; o > 0; o >>= 1) {
        if (threadIdx.x < o) red[threadIdx.x] += red[threadIdx.x + o];
        __syncthreads();
    }
    if (threadIdx.x == 0) costpart[blockIdx.x] = red[0];
}

__global__ void k_cost(const float* costpart, float* out0) {
    __shared__ float red[256];
    float s = 0.f;
    for (int t = threadIdx.x; t < 2048; t += 256) s += costpart[t];
    red[threadIdx.x] = s;
    __syncthreads();
    for (int o = 128; o > 0; o >>= 1) {
        if (threadIdx.x < o) red[threadIdx.x] += red[threadIdx.x + o];
        __syncthreads();
    }
    if (threadIdx.x == 0) out0[0] = red[0];
}

extern "C" void kernel_launch(void* const* d_in, const int* in_sizes, int n_in,
                              void* d_out, int out_size, void* d_ws, size_t ws_size,
                              hipStream_t stream) {
    const float* x  = (const float*)d_in[0];
    const float* y  = (const float*)d_in[1];
    const float* mu = (const float*)d_in[2];
    const float* nu = (const float*)d_in[3];

    float* out = (float*)d_out;
    float* P   = out + 1;                       // holds K during iterations
    float* C   = out + 1 + (size_t)NN * MMC;

    float* ws       = (float*)d_ws;
    float* colpart  = ws + WS_COLP;
    float* rowpart  = ws + WS_ROWP;
    float* disppart = ws + WS_DISPP;
    float* costpart = ws + WS_COSTP;
    int*   done     = (int*)(ws + WS_DONE);

    k_init<<<16, 256, 0, stream>>>(ws, out);
    k_cost_gemm<<<dim3(32, 256), 256, 0, stream>>>(x, y, C, P);

    for (int t = 0; t < MAX_ITER; ++t) {
        k_colsum<<<dim3(16, 32), 256, 0, stream>>>(P, ws, colpart, done);
        k_bnew<<<16, 256, 0, stream>>>(nu, ws, done);
        k_rowsum<<<dim3(4, 512), 256, 0, stream>>>(P, ws, rowpart, done);
        k_anew<<<16, 256, 0, stream>>>(mu, ws, done);
        k_disp<<<2048, 256, 0, stream>>>(P, ws, disppart, done, t == 0 ? 1 : 0);
        k_copy<<<16, 256, 0, stream>>>(ws, done);
        k_done<<<1, 256, 0, stream>>>(disppart, done);
    }

    k_final<<<2048, 256, 0, stream>>>(P, C, ws, costpart);
    k_cost<<<1, 256, 0, stream>>>(costpart, out);
}